// P2V_25941602468416
// MI455X (gfx1250) — compile-verified
//
#include <hip/hip_runtime.h>

typedef __attribute__((ext_vector_type(2))) float v2f;
typedef __attribute__((ext_vector_type(8))) float v8f;

#define B_TOT   4
#define G_TOT   32768   // D*H*W = 32*32*32
#define N_TOT   1024
#define LOG2E   1.44269504088896340736f

// One wave32 handles one (b, 16-grid-point) tile and reduces over all N=1024.
//
// dist[g,n] = x2[n] + g2[g] - 2 * sum_c xyz[b,c,n]*grid[c,g]
//
// We need t = -|s1|*dist*log2e for exp2, so fold k1 = -|s1|*log2e directly into
// the WMMA operands:  A = (-2*k1)*grid (K=c padded to 4),  B = xyz,
// C = k1*(x2[n] + g2[g])  ==>  D = k1*dist straight out of the accumulator.
// Then per element:  u = exp2(D) = exp(-|s1|dist);  ev = exp2(k2*u) = exp(v),
// with k2 = |s2|*log2e.  margin = (ev_top1 - ev_top2) / sum_n ev.
__global__ __launch_bounds__(256) void p2v_margin_kernel(
    const float* __restrict__ xyz,    // (B, 3, N)
    const float* __restrict__ grid,   // (3, D, H, W) -> (3, G_TOT)
    const float* __restrict__ s1p,    // (1,)
    const float* __restrict__ s2p,    // (1,)
    float* __restrict__ out)          // (B, D, H, W) -> (B, G_TOT)
{
  const int lane = threadIdx.x & 31;
  const int wave = blockIdx.x * (blockDim.x >> 5) + (threadIdx.x >> 5);
  const int b      = wave >> 11;           // wave / 2048
  const int g_base = (wave & 2047) << 4;   // 16 grid points per wave

  const float k1 = -fabsf(s1p[0]) * LOG2E;   // <= 0
  const float k2 =  fabsf(s2p[0]) * LOG2E;

  const int  m  = lane & 15;
  const bool hi = lane >= 16;
  const int  g_col = g_base + m;

  // ---- Loop-invariant A tile: lane m holds column g_base+m of grid ----
  const float gr0 = grid[0 * G_TOT + g_col];
  const float gr1 = grid[1 * G_TOT + g_col];
  const float gr2 = grid[2 * G_TOT + g_col];
  const float g2k_own = k1 * (gr0 * gr0 + gr1 * gr1 + gr2 * gr2);  // k1*g2[g]

  // A (16x4 f32): lanes 0-15 -> VGPR0=K0, VGPR1=K1 ; lanes 16-31 -> VGPR0=K2, VGPR1=K3(pad 0)
  const float m2k1 = -2.0f * k1;
  v2f a;
  a.x = hi ? (m2k1 * gr2) : (m2k1 * gr0);
  a.y = hi ? 0.0f         : (m2k1 * gr1);

  // k1*g2 for the 8 C/D rows this lane holds: row r -> g = g_base + r + (hi ? 8 : 0)
  float g2k[8];
#pragma unroll
  for (int r = 0; r < 8; ++r)
    g2k[r] = __shfl(g2k_own, hi ? (24 + r) : r, 32);

  float sumExp[8], t1[8], t2[8];
#pragma unroll
  for (int r = 0; r < 8; ++r) { sumExp[r] = 0.0f; t1[r] = -1.0f; t2[r] = -1.0f; }

  const float* xb = xyz + b * (3 * N_TOT);

  // ---- Stream N in 16-wide chunks: 1 WMMA + 8 exp2-pairs per lane per chunk ----
  for (int n0 = 0; n0 < N_TOT; n0 += 16) {
    const int n = n0 + m;
    const float x0  = xb[0 * N_TOT + n];
    const float x1  = xb[1 * N_TOT + n];
    const float x2c = xb[2 * N_TOT + n];
    const float xk  = k1 * (x0 * x0 + x1 * x1 + x2c * x2c);  // k1*x2[n]

    // B (4x16 f32): lanes 0-15 -> VGPR0=K0 row, VGPR1=K1 ; lanes 16-31 -> K2, K3(pad 0)
    v2f bm;
    bm.x = hi ? x2c  : x0;
    bm.y = hi ? 0.0f : x1;

    // C[row][n] = k1*(x2[n] + g2[g])
    v8f c;
#pragma unroll
    for (int r = 0; r < 8; ++r) c[r] = xk + g2k[r];

    // D = A*B + C  ==  k1 * dist[g, n]   (one 16x16 tile)
    v8f d = __builtin_amdgcn_wmma_f32_16x16x4_f32(
        /*neg_a=*/false, a, /*neg_b=*/false, bm,
        /*c_mod=*/(short)0, c, /*reuse_a=*/false, /*reuse_b=*/false);

#pragma unroll
    for (int r = 0; r < 8; ++r) {
      const float u  = __builtin_amdgcn_exp2f(d[r]);        // exp(-|s1|*dist)
      const float ev = __builtin_amdgcn_exp2f(k2 * u);      // exp(v)
      sumExp[r] += ev;
      const float lo = fminf(ev, t1[r]);                    // fuses to v_minmax
      t1[r] = fmaxf(ev, t1[r]);
      t2[r] = fmaxf(t2[r], lo);
    }
  }

  // ---- Half-wave (16-lane) reduction: each half shares the same 8 rows ----
#pragma unroll
  for (int ofs = 8; ofs >= 1; ofs >>= 1) {
#pragma unroll
    for (int r = 0; r < 8; ++r) {
      const float oS = __shfl_xor(sumExp[r], ofs, 32);
      const float o1 = __shfl_xor(t1[r],     ofs, 32);
      const float o2 = __shfl_xor(t2[r],     ofs, 32);
      sumExp[r] += oS;
      const float lo = fminf(t1[r], o1);
      t1[r] = fmaxf(t1[r], o1);
      t2[r] = fmaxf(fmaxf(t2[r], o2), lo);
    }
  }

  // margin = (exp(v_top1) - exp(v_top2)) / sum_n exp(v_n)
  if (m == 0) {
    const int gout = b * G_TOT + g_base + (hi ? 8 : 0);
#pragma unroll
    for (int r = 0; r < 8; ++r)
      out[gout + r] = (t1[r] - t2[r]) / sumExp[r];
  }
}

extern "C" void kernel_launch(void* const* d_in, const int* in_sizes, int n_in,
                              void* d_out, int out_size, void* d_ws, size_t ws_size,
                              hipStream_t stream) {
  (void)in_sizes; (void)n_in; (void)out_size; (void)d_ws; (void)ws_size;
  const float* xyz  = (const float*)d_in[0];
  const float* grid = (const float*)d_in[1];
  const float* s1   = (const float*)d_in[2];
  const float* s2   = (const float*)d_in[3];
  float* out = (float*)d_out;

  // 8192 waves = B(4) * G_TOT/16 (2048); 8 waves per 256-thread block -> 1024 blocks
  dim3 block(256);
  dim3 grd(1024);
  p2v_margin_kernel<<<grd, block, 0, stream>>>(xyz, grid, s1, s2, out);
}